// BackProjLayer_57045755625641
// MI455X (gfx1250) — compile-verified
//
#include <hip/hip_runtime.h>

typedef __attribute__((ext_vector_type(2))) float v2f;
typedef __attribute__((ext_vector_type(8))) float v8f;

#define N_SAMPLES 65536
#define N_PX      242
#define N_PX_PAD  256

// ---------------------------------------------------------------------------
// Phase 1: build W[p][k], p in [0,256), k in [0,32), column-major per pixel.
//   k in [0,16):  W[p][c*4+d]      = Dre[c,p]*Dre[d,p] + Dim[c,p]*Dim[d,p]
//   k in [16,32): W[p][16+c*4+d]   = Dim[c,p]*Dre[d,p] - Dre[c,p]*Dim[d,p]
// Pixels >= 242 are zero-padded so the padded GEMM columns are exact zeros.
// ---------------------------------------------------------------------------
__global__ void bp_prep_kernel(const float* __restrict__ Dre,
                               const float* __restrict__ Dim,
                               float* __restrict__ W) {
    const int p = threadIdx.x;            // 0..255, single block of 256
    float* wp = W + p * 32;
    if (p < N_PX) {
        float dre[4], dim_[4];
#pragma unroll
        for (int c = 0; c < 4; ++c) {
            dre[c]  = Dre[c * N_PX + p];
            dim_[c] = Dim[c * N_PX + p];
        }
#pragma unroll
        for (int c = 0; c < 4; ++c) {
#pragma unroll
            for (int d = 0; d < 4; ++d) {
                wp[c * 4 + d]      = dre[c] * dre[d]  + dim_[c] * dim_[d];
                wp[16 + c * 4 + d] = dim_[c] * dre[d] - dre[c]  * dim_[d];
            }
        }
    } else {
#pragma unroll
        for (int k = 0; k < 32; ++k) wp[k] = 0.0f;
    }
}

// ---------------------------------------------------------------------------
// Phase 2: out[n,p] = sum_k A[n][k] * W[p][k] - tau[p]
// A[n][k] = Sre[n*16+k] (k<16) else Sim[n*16+k-16].
// One wave computes a 16-row sample stripe; loops 16 pixel tiles of 16.
// V_WMMA_F32_16X16X4_F32, 8 K-steps of 4 per tile.
//
// VGPR layouts (ISA 7.12.2, 32-bit, wave32), half=lane>>4, m=lane&15:
//   A 16x4 : vgpr j = A[row m][k0 + 2*half + j]        -> float2 load
//   B 4x16 : vgpr j = B[k0 + 2*half + j][col m]        -> float2 load of W[col][..]
//   C/D    : vgpr r = D[row r + 8*half][col m]
// ---------------------------------------------------------------------------
__global__ __launch_bounds__(256) void bp_gemm_kernel(
        const float* __restrict__ Sre,
        const float* __restrict__ Sim,
        const float* __restrict__ W,
        const float* __restrict__ tau,
        float* __restrict__ out) {
    const int lane = threadIdx.x & 31;
    const int wave = threadIdx.x >> 5;
    const int half = lane >> 4;           // 0 or 1
    const int m    = lane & 15;           // row (A) / col (B,D) within tile
    const int n0   = (blockIdx.x * 8 + wave) * 16;   // sample-row base

    // Preload all A fragments for this wave's 16x32 stripe (8 K-steps).
    const float* rowre = Sre + (size_t)(n0 + m) * 16;
    const float* rowim = Sim + (size_t)(n0 + m) * 16;
    v2f a[8];
#pragma unroll
    for (int kk = 0; kk < 4; ++kk)
        a[kk] = *(const v2f*)(rowre + 4 * kk + 2 * half);
#pragma unroll
    for (int kk = 4; kk < 8; ++kk)
        a[kk] = *(const v2f*)(rowim + 4 * kk + 2 * half - 16);

    // Sweep the 16 pixel tiles (242 padded to 256).
    for (int nt = 0; nt < 16; ++nt) {
        const int col = nt * 16 + m;
        const float* wcol = W + col * 32 + 2 * half;   // W[col][k] column-major
        v8f acc = {0.f, 0.f, 0.f, 0.f, 0.f, 0.f, 0.f, 0.f};
#pragma unroll
        for (int kk = 0; kk < 8; ++kk) {
            v2f b = *(const v2f*)(wcol + 4 * kk);
            acc = __builtin_amdgcn_wmma_f32_16x16x4_f32(
                /*neg_a=*/false, a[kk], /*neg_b=*/false, b,
                /*c_mod=*/(short)0, acc, /*reuse_a=*/false, /*reuse_b=*/false);
        }
        if (col < N_PX) {
            const float t = tau[col];
#pragma unroll
            for (int r = 0; r < 8; ++r) {
                out[(size_t)(n0 + r + 8 * half) * N_PX + col] = acc[r] - t;
            }
        }
    }
}

// ---------------------------------------------------------------------------
// d_in order (setup_inputs): S_re (N,4,4) f32, S_im (N,4,4) f32,
//                            D_re (4,242) f32, D_im (4,242) f32, tau (242) f32
// d_out: (N,242) f32. d_ws: 256*32 floats for W (32 KB).
// ---------------------------------------------------------------------------
extern "C" void kernel_launch(void* const* d_in, const int* in_sizes, int n_in,
                              void* d_out, int out_size, void* d_ws, size_t ws_size,
                              hipStream_t stream) {
    const float* Sre = (const float*)d_in[0];
    const float* Sim = (const float*)d_in[1];
    const float* Dre = (const float*)d_in[2];
    const float* Dim = (const float*)d_in[3];
    const float* tau = (const float*)d_in[4];
    float* out = (float*)d_out;
    float* W   = (float*)d_ws;

    bp_prep_kernel<<<1, 256, 0, stream>>>(Dre, Dim, W);

    // 65536 samples / (8 waves * 16 rows) = 512 blocks
    bp_gemm_kernel<<<N_SAMPLES / 128, 256, 0, stream>>>(Sre, Sim, W, tau, out);
}